// ForwardPolicy_2954937499927
// MI455X (gfx1250) — compile-verified
//
#include <hip/hip_runtime.h>
#include <hip/hip_bf16.h>

// ---------------- problem constants ----------------
#define GSZ   48
#define NN    2304          // nodes per sample
#define BB    32            // batch
#define R2    0.09f         // radius^2
#define KNB   4             // max neighbors
#define HH    4             // heads
#define DD    32            // per-head dim
#define CC    128           // H*D
#define HID   4418          // 2*(GS-1)^2
#define KP    4432          // HID padded up to multiple of 16 (and 4)
#define NAOUT 2048

typedef __attribute__((ext_vector_type(2))) float v2f;
typedef __attribute__((ext_vector_type(8))) float v8f;

__device__ __forceinline__ float fixnan(float v) { return (v != v) ? -10.0f : v; }

// =====================================================================
// Kernel 1: radius graph + top-4 neighbors.  One thread per target node,
// sample coordinates staged once in LDS (18.4KB of the 320KB WGP pool).
// =====================================================================
__global__ __launch_bounds__(256)
void radius_topk(const float* __restrict__ data, int* __restrict__ nbr,
                 float* __restrict__ dist) {
  __shared__ float sx[NN * 2];
  const int b = blockIdx.y;
  const int tid = threadIdx.x;
  for (int t = tid; t < NN; t += 256) {
    sx[2 * t + 0] = fixnan(data[((size_t)b * NN + t) * 2 + 0]);
    sx[2 * t + 1] = fixnan(data[((size_t)b * NN + t) * 2 + 1]);
  }
  __syncthreads();
  const int i = blockIdx.x * 256 + tid;            // NN == 9*256, exact
  const float xi0 = sx[2 * i], xi1 = sx[2 * i + 1];
  float d0 = 3.0e38f, d1 = 3.0e38f, d2 = 3.0e38f, d3 = 3.0e38f;
  int   i0 = -1, i1 = -1, i2 = -1, i3 = -1;
  for (int j = 0; j < NN; ++j) {
    float dx = xi0 - sx[2 * j];
    float dy = xi1 - sx[2 * j + 1];
    float dd = fmaf(dy, dy, dx * dx);
    if (dd < R2 && j != i && dd < d3) {            // sorted insert (stable on ties)
      if (dd < d2) {
        d3 = d2; i3 = i2;
        if (dd < d1) {
          d2 = d1; i2 = i1;
          if (dd < d0) { d1 = d0; i1 = i0; d0 = dd; i0 = j; }
          else         { d1 = dd; i1 = j; }
        } else { d2 = dd; i2 = j; }
      } else { d3 = dd; i3 = j; }
    }
  }
  const size_t base = ((size_t)b * NN + i) * KNB;
  nbr[base + 0] = i0;  dist[base + 0] = (i0 >= 0) ? sqrtf(d0) : 0.0f;
  nbr[base + 1] = i1;  dist[base + 1] = (i1 >= 0) ? sqrtf(d1) : 0.0f;
  nbr[base + 2] = i2;  dist[base + 2] = (i2 >= 0) ? sqrtf(d2) : 0.0f;
  nbr[base + 3] = i3;  dist[base + 3] = (i3 >= 0) ? sqrtf(d3) : 0.0f;
}

// =====================================================================
// Kernel 2: TransformerConv.  One wave32 per node.  Lane l owns channels
// c = h*32 + l (h = 0..3) so each head's q.k dot product is a full 32-lane
// butterfly reduction; 4 heads reduced simultaneously.
// =====================================================================
__global__ __launch_bounds__(128)
void attn_node(const float* __restrict__ data, const int* __restrict__ nbr,
               const float* __restrict__ dist,
               const float* __restrict__ Wq, const float* __restrict__ bq,
               const float* __restrict__ Wk, const float* __restrict__ bk,
               const float* __restrict__ Wv, const float* __restrict__ bv,
               const float* __restrict__ We,
               const float* __restrict__ Ws, const float* __restrict__ bs,
               float* __restrict__ outbuf) {
  const int w    = blockIdx.x * 4 + (threadIdx.x >> 5);  // global wave id == node id
  const int lane = threadIdx.x & 31;
  const int b = w / NN, n = w % NN;

  const float xi0 = fixnan(data[((size_t)b * NN + n) * 2 + 0]);
  const float xi1 = fixnan(data[((size_t)b * NN + n) * 2 + 1]);

  float q[HH];
#pragma unroll
  for (int h = 0; h < HH; ++h) {
    const int c = h * 32 + lane;
    q[h] = fmaf(xi0, Wq[c], fmaf(xi1, Wq[CC + c], bq[c]));
  }

  float vcache[KNB][HH];
  float logit[KNB][HH];
  const size_t ebase = ((size_t)b * NN + n) * KNB;
#pragma unroll
  for (int kk = 0; kk < KNB; ++kk) {
    const int j  = nbr[ebase + kk];
    const bool ok = (j >= 0);
    const int jc = ok ? j : 0;
    const float dj = dist[ebase + kk];
    const float xj0 = fixnan(data[((size_t)b * NN + jc) * 2 + 0]);
    const float xj1 = fixnan(data[((size_t)b * NN + jc) * 2 + 1]);
    float p[HH];
#pragma unroll
    for (int h = 0; h < HH; ++h) {
      const int c = h * 32 + lane;
      const float e  = dj * We[c];
      const float kv = fmaf(xj0, Wk[c], fmaf(xj1, Wk[CC + c], bk[c])) + e;
      const float vv = fmaf(xj0, Wv[c], fmaf(xj1, Wv[CC + c], bv[c])) + e;
      vcache[kk][h] = vv;
      p[h] = q[h] * kv;
    }
#pragma unroll
    for (int off = 16; off > 0; off >>= 1) {       // 4 parallel 32-lane reductions
#pragma unroll
      for (int h = 0; h < HH; ++h) p[h] += __shfl_xor(p[h], off, 32);
    }
    const float inv_sqrtD = 0.17677669529663689f;  // 1/sqrt(32)
#pragma unroll
    for (int h = 0; h < HH; ++h) logit[kk][h] = ok ? p[h] * inv_sqrtD : -1.0e30f;
  }

  // per-head softmax over the (<=4) neighbors, then weighted sum + skip + relu
#pragma unroll
  for (int h = 0; h < HH; ++h) {
    float m = logit[0][h];
#pragma unroll
    for (int kk = 1; kk < KNB; ++kk) m = fmaxf(m, logit[kk][h]);
    float wv[KNB], s = 0.0f;
#pragma unroll
    for (int kk = 0; kk < KNB; ++kk) {
      wv[kk] = (logit[kk][h] > -1.0e29f) ? __expf(logit[kk][h] - m) : 0.0f;
      s += wv[kk];
    }
    const float inv = 1.0f / fmaxf(s, 1.0e-12f);
    float acc = 0.0f;
#pragma unroll
    for (int kk = 0; kk < KNB; ++kk) acc = fmaf(wv[kk] * inv, vcache[kk][h], acc);
    const int c = h * 32 + lane;
    const float skip = fmaf(xi0, Ws[c], fmaf(xi1, Ws[CC + c], bs[c]));
    outbuf[((size_t)b * NN + n) * CC + c] = fmaxf(acc + skip, 0.0f);
  }
}

// =====================================================================
// Kernel 3: 2x2 VALID conv over raw .view(C,48,48); weights in LDS.
// Writes conv output zero-padded to KP columns (branch-free GEMM A tail).
// =====================================================================
__global__ __launch_bounds__(256)
void conv2x2(const float* __restrict__ outbuf, const float* __restrict__ convW,
             const float* __restrict__ convb, float* __restrict__ convout) {
  __shared__ float sw[2 * CC * 4];                 // 1024 floats
  const int b = blockIdx.x, tid = threadIdx.x;
  for (int t = tid; t < 2 * CC * 4; t += 256) sw[t] = convW[t];
  __syncthreads();
  const int idx = blockIdx.y * 256 + tid;
  if (idx >= KP) return;
  float* drow = convout + (size_t)b * KP;
  if (idx >= HID) { drow[idx] = 0.0f; return; }    // zero pad
  const int o = idx / 2209;
  const int r = idx - o * 2209;
  const int y = r / 47;
  const int x = r - y * 47;
  const float* img = outbuf + (size_t)b * NN * CC; // flat (C,48,48) reinterpretation
  float acc = convb[o];
#pragma unroll 4
  for (int c = 0; c < CC; ++c) {
    const int base = c * (GSZ * GSZ) + y * GSZ + x;
    const float* wc = &sw[(o * CC + c) * 4];
    acc = fmaf(img[base],           wc[0], acc);
    acc = fmaf(img[base + 1],       wc[1], acc);
    acc = fmaf(img[base + GSZ],     wc[2], acc);
    acc = fmaf(img[base + GSZ + 1], wc[3], acc);
  }
  drow[idx] = acc;
}

// =====================================================================
// Kernel 4: skinny f32 WMMA GEMM.  One wave computes a 32x16 output tile
// (two V_WMMA_F32_16X16X4_F32 per k-step sharing one B fragment -> each
// weight of W is read exactly once across the grid).  grid.y = split-K.
// All shape parameters are compile-time so B-side addressing strength-
// reduces to pointer += constant, with the second K-row and the 32-row
// prefetch as constant immediate offsets (both < 24-bit IOFFSET).
//   A: (32, KP) row-major, zero-padded cols [K_, KP)
//   B: (K_, LDB_) row-major weights
//   part: [split][32][NP_] partial sums
// ISA fragment layouts (CDNA5 ISA):
//   A 16x4: lanes 0-15 rows M, VGPRs {K+0,K+1}; lanes 16-31 {K+2,K+3}
//   B 4x16: VGPR0 = rows {K+0 (lanes0-15), K+1 (lanes16-31)}, VGPR1 = {K+2,K+3}
//   D:      VGPR p -> M = 8*half + p, N = lane&15
// =====================================================================
template <int K_, int NOUT_, int LDB_, int NP_, int KCHUNK_>
__global__ __launch_bounds__(32)
void gemm32_wmma(const float* __restrict__ A, const float* __restrict__ Bm,
                 float* __restrict__ part) {
  const int nt   = blockIdx.x;
  const int sk   = blockIdx.y;
  const int lane = threadIdx.x;
  const int half = lane >> 4;
  const int lid  = lane & 15;
  const int n    = nt * 16 + lid;
  const int ncl  = (n < NOUT_) ? n : 0;            // clamped column (no EXEC divergence)
  const int k0 = sk * KCHUNK_;
  int k1 = k0 + KCHUNK_; if (k1 > K_) k1 = K_;

  v8f acc0 = {}, acc1 = {};
  const float* arow0 = A + (size_t)lid * KP;       // M-tile 0 (rows 0..15)
  const float* arow1 = A + (size_t)(16 + lid) * KP;// M-tile 1 (rows 16..31)
  const float* bp    = Bm + (size_t)(k0 + half) * LDB_ + ncl;  // row k+half

  int k = k0;
  // ---- main phase with unconditional constant-offset prefetch (stops 32 rows early)
#pragma unroll 4
  for (; k + 36 <= k1; k += 4) {
    __builtin_prefetch(bp + 32 * LDB_, 0, 1);      // global_prefetch_b8, imm offset
    const float2 a0 = *(const float2*)(arow0 + k + 2 * half);
    const float2 a1 = *(const float2*)(arow1 + k + 2 * half);
    const float bx = bp[0];                        // row k+half
    const float by = bp[2 * LDB_];                 // row k+2+half (imm offset)
    const v2f av0 = {a0.x, a0.y};
    const v2f av1 = {a1.x, a1.y};
    const v2f bv  = {bx, by};
    acc0 = __builtin_amdgcn_wmma_f32_16x16x4_f32(false, av0, false, bv, (short)0, acc0, false, false);
    acc1 = __builtin_amdgcn_wmma_f32_16x16x4_f32(false, av1, false, bv, (short)0, acc1, false, false);
    bp += 4 * LDB_;
  }
  // ---- drain phase (last 8 k-steps), no prefetch
#pragma unroll 4
  for (; k + 4 <= k1; k += 4) {
    const float2 a0 = *(const float2*)(arow0 + k + 2 * half);
    const float2 a1 = *(const float2*)(arow1 + k + 2 * half);
    const float bx = bp[0];
    const float by = bp[2 * LDB_];
    const v2f av0 = {a0.x, a0.y};
    const v2f av1 = {a1.x, a1.y};
    const v2f bv  = {bx, by};
    acc0 = __builtin_amdgcn_wmma_f32_16x16x4_f32(false, av0, false, bv, (short)0, acc0, false, false);
    acc1 = __builtin_amdgcn_wmma_f32_16x16x4_f32(false, av1, false, bv, (short)0, acc1, false, false);
    bp += 4 * LDB_;
  }
  if (k < k1) {                                    // K tail (<=3), A is zero-padded
    const float2 a0 = *(const float2*)(arow0 + k + 2 * half);
    const float2 a1 = *(const float2*)(arow1 + k + 2 * half);
    const int r0 = k + half, r1 = k + 2 + half;
    const int r0c = (r0 < K_) ? r0 : (K_ - 1);
    const int r1c = (r1 < K_) ? r1 : (K_ - 1);
    float bx = Bm[(size_t)r0c * LDB_ + ncl];
    float by = Bm[(size_t)r1c * LDB_ + ncl];
    bx = (r0 < k1) ? bx : 0.0f;                    // v_cndmask, EXEC stays all-ones
    by = (r1 < k1) ? by : 0.0f;
    const v2f av0 = {a0.x, a0.y};
    const v2f av1 = {a1.x, a1.y};
    const v2f bv  = {bx, by};
    acc0 = __builtin_amdgcn_wmma_f32_16x16x4_f32(false, av0, false, bv, (short)0, acc0, false, false);
    acc1 = __builtin_amdgcn_wmma_f32_16x16x4_f32(false, av1, false, bv, (short)0, acc1, false, false);
  }

  if (n < NOUT_) {
    float* p0 = part + (size_t)sk * 32 * NP_;
#pragma unroll
    for (int p = 0; p < 8; ++p) {
      const int m = half * 8 + p;                  // D layout: VGPR p -> row 8*half+p
      p0[(size_t)m * NP_ + n]        = acc0[p];
      p0[(size_t)(16 + m) * NP_ + n] = acc1[p];
    }
  }
}

// =====================================================================
// Kernel 5: split-K reduction + bias (+ optional ReLU), zero-pads dest.
// =====================================================================
__global__ __launch_bounds__(256)
void reduce_bias(const float* __restrict__ part, const float* __restrict__ bias,
                 float* __restrict__ dst, int splits, int Nout, int ldd, int relu) {
  const int t = blockIdx.x * 256 + threadIdx.x;    // over 32*ldd
  const int m = t / ldd, n = t - m * ldd;
  if (n >= Nout) { dst[(size_t)m * ldd + n] = 0.0f; return; }
  float s = bias[n];
  for (int sk = 0; sk < splits; ++sk) s += part[((size_t)sk * 32 + m) * ldd + n];
  dst[(size_t)m * ldd + n] = relu ? fmaxf(s, 0.0f) : s;
}

// =====================================================================
// Kernel 6: row softmax (32 rows x 2048).
// =====================================================================
__global__ __launch_bounds__(256)
void softmax_rows(const float* __restrict__ logits, float* __restrict__ out) {
  __shared__ float red[256];
  const int row = blockIdx.x, tid = threadIdx.x;
  const float* lr = logits + (size_t)row * NAOUT;
  float v[8], m = -3.0e38f;
#pragma unroll
  for (int t = 0; t < 8; ++t) { v[t] = lr[tid + t * 256]; m = fmaxf(m, v[t]); }
  red[tid] = m; __syncthreads();
  for (int s = 128; s > 0; s >>= 1) {
    if (tid < s) red[tid] = fmaxf(red[tid], red[tid + s]);
    __syncthreads();
  }
  m = red[0]; __syncthreads();
  float s = 0.0f;
#pragma unroll
  for (int t = 0; t < 8; ++t) { v[t] = __expf(v[t] - m); s += v[t]; }
  red[tid] = s; __syncthreads();
  for (int s2 = 128; s2 > 0; s2 >>= 1) {
    if (tid < s2) red[tid] += red[tid + s2];
    __syncthreads();
  }
  const float inv = 1.0f / red[0];
#pragma unroll
  for (int t = 0; t < 8; ++t) out[(size_t)row * NAOUT + tid + t * 256] = v[t] * inv;
}

// =====================================================================
extern "C" void kernel_launch(void* const* d_in, const int* in_sizes, int n_in,
                              void* d_out, int out_size, void* d_ws, size_t ws_size,
                              hipStream_t stream) {
  const float* data  = (const float*)d_in[0];
  const float* Wq    = (const float*)d_in[1];
  const float* bq    = (const float*)d_in[2];
  const float* Wk    = (const float*)d_in[3];
  const float* bk    = (const float*)d_in[4];
  const float* Wv    = (const float*)d_in[5];
  const float* bv    = (const float*)d_in[6];
  const float* We    = (const float*)d_in[7];
  const float* Ws    = (const float*)d_in[8];
  const float* bs    = (const float*)d_in[9];
  const float* convW = (const float*)d_in[10];
  const float* convb = (const float*)d_in[11];
  const float* W1    = (const float*)d_in[12];
  const float* b1    = (const float*)d_in[13];
  const float* W2    = (const float*)d_in[14];
  const float* b2    = (const float*)d_in[15];

  char* ws = (char*)d_ws;
  size_t off = 0;
  auto alloc = [&](size_t bytes) -> void* {
    void* p = ws + off;
    off = (off + bytes + 255) & ~(size_t)255;
    return p;
  };
  int*   nbr     = (int*)  alloc((size_t)BB * NN * KNB * sizeof(int));   // 1.2 MB
  float* dist    = (float*)alloc((size_t)BB * NN * KNB * sizeof(float)); // 1.2 MB
  float* outbuf  = (float*)alloc((size_t)BB * NN * CC * sizeof(float));  // 37.7 MB
  float* convout = (float*)alloc((size_t)BB * KP * sizeof(float));       // 0.57 MB
  float* hbuf    = (float*)alloc((size_t)BB * KP * sizeof(float));       // 0.57 MB
  float* logitsb = (float*)alloc((size_t)BB * NAOUT * sizeof(float));    // 0.26 MB
  float* part1   = (float*)alloc((size_t)4 * BB * KP * sizeof(float));   // 2.27 MB
  float* part2   = (float*)alloc((size_t)8 * BB * NAOUT * sizeof(float));// 2.1 MB
  (void)ws_size; (void)in_sizes; (void)n_in; (void)out_size;

  // 1) radius graph + top-4
  radius_topk<<<dim3(NN / 256, BB), dim3(256), 0, stream>>>(data, nbr, dist);
  // 2) transformer conv -> (B, N, 128)
  attn_node<<<dim3((BB * NN) / 4), dim3(128), 0, stream>>>(
      data, nbr, dist, Wq, bq, Wk, bk, Wv, bv, We, Ws, bs, outbuf);
  // 3) 2x2 conv on .view(C,48,48) -> (B, KP) zero-padded
  conv2x2<<<dim3(BB, (KP + 255) / 256), dim3(256), 0, stream>>>(
      outbuf, convW, convb, convout);
  // 4) layer 1: (32 x 4418) @ (4418 x 4418), split-K=4
  gemm32_wmma<HID, HID, HID, KP, 1108><<<dim3((HID + 15) / 16, 4), dim3(32), 0, stream>>>(
      convout, W1, part1);
  reduce_bias<<<dim3((BB * KP) / 256), dim3(256), 0, stream>>>(
      part1, b1, hbuf, 4, HID, KP, 1);
  // 5) layer 2: (32 x 4418) @ (4418 x 2048), split-K=8
  gemm32_wmma<HID, NAOUT, NAOUT, NAOUT, 556><<<dim3(NAOUT / 16, 8), dim3(32), 0, stream>>>(
      hbuf, W2, part2);
  reduce_bias<<<dim3((BB * NAOUT) / 256), dim3(256), 0, stream>>>(
      part2, b2, logitsb, 8, NAOUT, NAOUT, 0);
  // 6) softmax -> d_out (32 x 2048)
  softmax_rows<<<dim3(BB), dim3(256), 0, stream>>>(logitsb, (float*)d_out);
}